// SimilarityOutputLayer_25967372271701
// MI455X (gfx1250) — compile-verified
//
#include <hip/hip_runtime.h>
#include <hip/hip_bf16.h>
#include <math.h>

typedef float v2f __attribute__((ext_vector_type(2)));
typedef float v8f __attribute__((ext_vector_type(8)));

#define EPS 1e-7f
#define NB   256   // batches
#define NU   8     // user embeddings per batch
#define NI   1024  // items per batch
#define ED   64    // embedding dim

// One wave -> 16(items) x 16(users; 8 valid + 8 zero pad) output tile,
// K=64 accumulated as 16 chained V_WMMA_F32_16X16X4_F32.
// Block = 256 threads = 8 waves -> 128 items. Grid = 256 * (1024/128) = 2048.
__global__ __launch_bounds__(256)
void simout_kernel(const float* __restrict__ user_emb,
                   const float* __restrict__ item_emb,
                   const float* __restrict__ proj_w,
                   const float* __restrict__ proj_b,
                   float* __restrict__ out)
{
    const int tid  = threadIdx.x;
    const int wave = tid >> 5;                 // 0..7
    const int lane = tid & 31;
    const int b    = blockIdx.x >> 3;          // batch
    const int blk  = blockIdx.x & 7;           // 128-item block
    const int m0   = blk * 128 + wave * 16;    // first item row of this wave

    const int sel = lane >> 4;                 // K-pair selector {0,1}
    const int l15 = lane & 15;                 // M for A-frag / N for B-frag

    // ---- A fragments: 16 item rows, normalize in registers ----
    const float* irow = item_emb + ((size_t)b * NI + (size_t)(m0 + l15)) * ED;
    v2f a[16];
    float ss = 0.0f;
#pragma unroll
    for (int kc = 0; kc < 16; ++kc) {
        const int k = kc * 4 + sel * 2;
        float x = irow[k];
        float y = irow[k + 1];
        a[kc].x = x; a[kc].y = y;
        ss = fmaf(x, x, fmaf(y, y, ss));
    }
    ss += __shfl_xor(ss, 16);                  // lanes L and L+16 hold the two row halves
    const float ascale = 1.0f / fmaxf(sqrtf(ss), EPS);
#pragma unroll
    for (int kc = 0; kc < 16; ++kc) { a[kc].x *= ascale; a[kc].y *= ascale; }

    // ---- B fragments: 8 user rows (pad N=8..15 with zeros, no divergence) ----
    const float* urow = user_emb + ((size_t)b * NU + (size_t)(l15 & 7)) * ED;
    const float uvalid = (l15 < NU) ? 1.0f : 0.0f;
    v2f bf[16];
    float us = 0.0f;
#pragma unroll
    for (int kc = 0; kc < 16; ++kc) {
        const int k = kc * 4 + sel * 2;
        float x = urow[k] * uvalid;
        float y = urow[k + 1] * uvalid;
        bf[kc].x = x; bf[kc].y = y;
        us = fmaf(x, x, fmaf(y, y, us));
    }
    us += __shfl_xor(us, 16);
    const float bscale = 1.0f / fmaxf(sqrtf(us), EPS);
#pragma unroll
    for (int kc = 0; kc < 16; ++kc) { bf[kc].x *= bscale; bf[kc].y *= bscale; }

    // ---- GEMM tile: 16 chained f32 WMMAs (EXEC is all-1s here) ----
    v8f c = {};
#pragma unroll
    for (int kc = 0; kc < 16; ++kc) {
        c = __builtin_amdgcn_wmma_f32_16x16x4_f32(
                /*neg_a=*/false, a[kc], /*neg_b=*/false, bf[kc],
                /*c_mod=*/(short)0, c, /*reuse_a=*/false, /*reuse_b=*/false);
    }

    // ---- epilogue: Linear(1,1) + sigmoid, store valid N (coalesced 32B runs) ----
    const float w  = proj_w[0];
    const float bb = proj_b[0];
    if (l15 < NU) {
        const int n = l15;
#pragma unroll
        for (int r = 0; r < 8; ++r) {
            const int item = m0 + r + sel * 8;     // C layout: lanes>=16 hold M=r+8
            float v = fmaf(c[r], w, bb);
            float s = 1.0f / (1.0f + __expf(-v));
            out[((size_t)b * NI + (size_t)item) * NU + n] = s;
        }
    }
}

extern "C" void kernel_launch(void* const* d_in, const int* in_sizes, int n_in,
                              void* d_out, int out_size, void* d_ws, size_t ws_size,
                              hipStream_t stream) {
    const float* user_emb = (const float*)d_in[0];  // (256, 8, 64)
    const float* item_emb = (const float*)d_in[1];  // (256, 1024, 64)
    const float* proj_w   = (const float*)d_in[2];  // (1, 1)
    const float* proj_b   = (const float*)d_in[3];  // (1,)
    float* out = (float*)d_out;                     // (256, 1024, 8)

    dim3 block(256);
    dim3 grid(NB * (NI / 128));                     // 2048 workgroups
    simout_kernel<<<grid, block, 0, stream>>>(user_emb, item_emb, proj_w, proj_b, out);
}